// CA_Drop_48352741818433
// MI455X (gfx1250) — compile-verified
//
#include <hip/hip_runtime.h>

typedef __attribute__((ext_vector_type(2))) float v2f;
typedef __attribute__((ext_vector_type(4))) float f4;
typedef __attribute__((ext_vector_type(8))) float v8f;

#define NM_   64
#define C_    256
#define T_    128
#define V_    25
#define TV_   (T_ * V_)        // 3200
#define CTV_  (C_ * TV_)       // 819200
#define NTOT_ (NM_ * CTV_)     // 52428800

// workspace layout (floats)
#define MK_S_OFF  0            // 1600 floats: mk_s[n*25+v]
#define MK_T_OFF  1600         // 8192 floats: mk_t[n*128+t]
#define SCALE_OFF 9792         // 1 float: combined scale
#define S_OFF     9824         // 204800 floats: fused s[n*3200 + t*25 + v] (16B aligned: 9824*4 % 16 == 0)

// ---------------------------------------------------------------------------
// Kernel 1: one block computes masks, the M_seed @ A matmul via WMMA f32
// 16x16x4, sliding-window max, and the global scale factors.
// ---------------------------------------------------------------------------
__global__ __launch_bounds__(1024) void ca_drop_setup(
    const float* __restrict__ A,
    const float* __restrict__ mask_s,
    const float* __restrict__ mask_t,
    const float* __restrict__ u_s,
    const float* __restrict__ u_t,
    float* __restrict__ ws)
{
    __shared__ float red[1024];
    __shared__ float seed[64 * 28];   // M_seed padded K: 25 -> 28
    __shared__ float amat[28 * 32];   // A padded to 28 x 32 (zeros)
    __shared__ float mks[NM_ * V_];   // mk_s
    __shared__ float mtb[NM_ * T_];   // Mt
    __shared__ float s_sum_ms, s_sum_mt, s_sum_mks, s_sum_mkt;

    const int tid = threadIdx.x;

    // ---- sum(mask_s) over 1600 ----
    float acc = 0.0f;
    for (int i = tid; i < NM_ * V_; i += 1024) acc += mask_s[i];
    red[tid] = acc; __syncthreads();
    for (int s2 = 512; s2 > 0; s2 >>= 1) {
        if (tid < s2) red[tid] += red[tid + s2];
        __syncthreads();
    }
    if (tid == 0) s_sum_ms = red[0];
    __syncthreads();

    // ---- sum(mask_t) over 8192 ----
    acc = 0.0f;
    for (int i = tid; i < NM_ * T_; i += 1024) acc += mask_t[i];
    red[tid] = acc; __syncthreads();
    for (int s2 = 512; s2 > 0; s2 >>= 1) {
        if (tid < s2) red[tid] += red[tid + s2];
        __syncthreads();
    }
    if (tid == 0) s_sum_mt = red[0];
    __syncthreads();

    // ---- M_seed into LDS (zero-padded 64x28), A into LDS (28x32) ----
    const float inv_ms  = 1600.0f / s_sum_ms;
    const float gamma_s = (1.0f - 0.9f) / (1.0f + 1.92f);
    for (int i = tid; i < 64 * 28; i += 1024) {
        int n = i / 28, k = i - n * 28;
        float v = 0.0f;
        if (k < V_) {
            float p = fminf(mask_s[n * V_ + k] * inv_ms * gamma_s, 1.0f);
            v = (u_s[n * V_ + k] < p) ? 1.0f : 0.0f;
        }
        seed[i] = v;
    }
    for (int i = tid; i < 28 * 32; i += 1024) {
        int k = i / 32, v = i - k * 32;
        amat[i] = (k < V_ && v < V_) ? A[k * V_ + v] : 0.0f;
    }
    __syncthreads();

    // ---- rowsum = seed(64x28) @ amat(28x32) with V_WMMA_F32_16X16X4_F32 ----
    // 8 waves, each owns one 16x16 tile: mtile = wave/2, ntile = wave%2.
    const int wave = tid >> 5;
    const int lane = tid & 31;
    if (wave < 8) {
        const int mtile = wave >> 1;
        const int ntile = wave & 1;
        const int mrow  = mtile * 16 + (lane & 15);
        const int ncol  = ntile * 16 + (lane & 15);
        const int khalf = (lane < 16) ? 0 : 2;     // A 16x4 layout: lanes 0-15 K0/K1, 16-31 K2/K3
        v8f accv = {};
        for (int k0 = 0; k0 < 28; k0 += 4) {
            v2f a, b;
            a.x = seed[mrow * 28 + k0 + khalf];
            a.y = seed[mrow * 28 + k0 + khalf + 1];
            b.x = amat[(k0 + khalf) * 32 + ncol];
            b.y = amat[(k0 + khalf + 1) * 32 + ncol];
            accv = __builtin_amdgcn_wmma_f32_16x16x4_f32(
                false, a, false, b, (short)0, accv, false, false);
        }
        // C/D layout: VGPR r holds M = r (lanes 0-15) or M = r+8 (lanes 16-31), N = lane%16
        const int v = ntile * 16 + (lane & 15);
        if (v < V_) {
            const int nbase = mtile * 16 + ((lane < 16) ? 0 : 8);
#pragma unroll
            for (int r = 0; r < 8; ++r) {
                float mk = (accv[r] > 0.001f) ? 0.0f : 1.0f;   // mk_s = 1 - (M_seed@A > 0.001)
                mks[(nbase + r) * V_ + v] = mk;
            }
        }
    }
    __syncthreads();

    // ---- write mk_s to ws + sum ----
    acc = 0.0f;
    for (int i = tid; i < NM_ * V_; i += 1024) {
        float m = mks[i];
        ws[MK_S_OFF + i] = m;
        acc += m;
    }
    red[tid] = acc; __syncthreads();
    for (int s2 = 512; s2 > 0; s2 >>= 1) {
        if (tid < s2) red[tid] += red[tid + s2];
        __syncthreads();
    }
    if (tid == 0) s_sum_mks = red[0];
    __syncthreads();

    // ---- Mt, then sliding-window max width 7 (pad 3), mk_t = 1 - max ----
    const float inv_mt  = 8192.0f / s_sum_mt;
    const float gamma_t = (1.0f - 0.9f) / 7.0f;
    for (int i = tid; i < NM_ * T_; i += 1024) {
        float p = fminf(mask_t[i] * inv_mt * gamma_t, 1.0f);
        mtb[i] = (u_t[i] < p) ? 1.0f : 0.0f;
    }
    __syncthreads();
    acc = 0.0f;
    for (int i = tid; i < NM_ * T_; i += 1024) {
        int n = i >> 7, t = i & 127;
        int lo = (t - 3 < 0) ? 0 : t - 3;
        int hi = (t + 3 > 127) ? 127 : t + 3;
        float m = 0.0f;
        for (int tt = lo; tt <= hi; ++tt) m = fmaxf(m, mtb[n * T_ + tt]);
        float mk = 1.0f - m;
        ws[MK_T_OFF + i] = mk;
        acc += mk;
    }
    red[tid] = acc; __syncthreads();
    for (int s2 = 512; s2 > 0; s2 >>= 1) {
        if (tid < s2) red[tid] += red[tid + s2];
        __syncthreads();
    }
    if (tid == 0) {
        s_sum_mkt = red[0];
        float scale_s = (float)(NM_ * V_) / fmaxf(s_sum_mks, 1.0f);
        float scale_t = (float)NTOT_ / fmaxf(s_sum_mkt * (float)(C_ * V_), 1.0f);
        ws[SCALE_OFF] = scale_s * scale_t;
    }
}

// ---------------------------------------------------------------------------
// Kernel 2: build fused per-(n,t,v) scale table s[n*3200 + t*25 + v]
// (800 KB, stays resident in the 192 MB L2; reused C=256 times)
// ---------------------------------------------------------------------------
__global__ __launch_bounds__(256) void ca_drop_stable(float* __restrict__ ws)
{
    const int i = blockIdx.x * 256 + threadIdx.x;   // < 204800
    const int n = i / TV_;
    const int r = i - n * TV_;
    const int t = r / V_;
    const int v = r - t * V_;
    const float scale = ws[SCALE_OFF];
    ws[S_OFF + i] = ws[MK_S_OFF + n * V_ + v] * ws[MK_T_OFF + n * T_ + t] * scale;
}

// ---------------------------------------------------------------------------
// Kernel 3: bandwidth kernel.  out = x * s   (B128 NT load / NT store;
// s-table load hits L2).  Exact: i % (T*V) == t*V+v, i / (C*T*V) == n.
// ---------------------------------------------------------------------------
__global__ __launch_bounds__(256) void ca_drop_scale(
    const f4* __restrict__ x4,
    const float* __restrict__ stab,
    f4* __restrict__ out4)
{
    const unsigned i4   = blockIdx.x * 256u + threadIdx.x;   // < 13,107,200
    const unsigned base = i4 * 4u;
    const unsigned n    = base / (unsigned)CTV_;
    const unsigned r    = base % (unsigned)TV_;              // 4-aligned
    f4 xv = __builtin_nontemporal_load(&x4[i4]);
    f4 sv = *(const f4*)(stab + (size_t)n * TV_ + r);
    __builtin_nontemporal_store(xv * sv, &out4[i4]);
}

// Fallback if workspace is too small for the s-table: compute scale inline.
__global__ __launch_bounds__(256) void ca_drop_scale_direct(
    const f4* __restrict__ x4,
    const float* __restrict__ ws,
    f4* __restrict__ out4)
{
    const unsigned i4   = blockIdx.x * 256u + threadIdx.x;
    const unsigned base = i4 * 4u;
    const unsigned n    = base / (unsigned)CTV_;
    const unsigned r    = base % (unsigned)TV_;
    const float scale = ws[SCALE_OFF];
    f4 xv = __builtin_nontemporal_load(&x4[i4]);
    f4 o;
#pragma unroll
    for (int j = 0; j < 4; ++j) {
        unsigned rr = r + j;                 // stays within same n (TV_ % 4 == 0)
        unsigned t = rr / V_, v = rr - t * V_;
        float s = ws[MK_S_OFF + n * V_ + v] * ws[MK_T_OFF + n * T_ + t] * scale;
        o[j] = xv[j] * s;
    }
    __builtin_nontemporal_store(o, &out4[i4]);
}

extern "C" void kernel_launch(void* const* d_in, const int* in_sizes, int n_in,
                              void* d_out, int out_size, void* d_ws, size_t ws_size,
                              hipStream_t stream) {
    const float* x      = (const float*)d_in[0];
    const float* A      = (const float*)d_in[1];
    const float* mask_s = (const float*)d_in[2];
    const float* mask_t = (const float*)d_in[3];
    const float* u_s    = (const float*)d_in[4];
    const float* u_t    = (const float*)d_in[5];
    float* ws  = (float*)d_ws;
    float* out = (float*)d_out;

    ca_drop_setup<<<1, 1024, 0, stream>>>(A, mask_s, mask_t, u_s, u_t, ws);

    const size_t need = (size_t)(S_OFF + (size_t)NM_ * TV_) * sizeof(float);
    const int nvec = NTOT_ / 4;            // 13,107,200 float4s
    if (ws_size >= need) {
        ca_drop_stable<<<(NM_ * TV_) / 256, 256, 0, stream>>>(ws);
        ca_drop_scale<<<nvec / 256, 256, 0, stream>>>(
            (const f4*)x, ws + S_OFF, (f4*)out);
    } else {
        ca_drop_scale_direct<<<nvec / 256, 256, 0, stream>>>(
            (const f4*)x, ws, (f4*)out);
    }
}